// DenseNGCNLayer_57097295233453
// MI455X (gfx1250) — compile-verified
//
#include <hip/hip_runtime.h>

typedef __attribute__((ext_vector_type(2))) float v2f;
typedef __attribute__((ext_vector_type(8))) float v8f;

#define IN_CH 512
#define OUT_CH 64
#define KC 128            // K rows of weight staged in LDS per chunk
#define TS  66            // transposed LDS row stride in dwords (64 data + 2 pad)
#define HALF_SZ (OUT_CH * TS)         // dwords per half-region (4224)
#define HBASE0 0
#define HBASE1 (HALF_SZ + 32)         // +32 dwords -> second half-wave hits banks +32

// ---------------------------------------------------------------------------
// Dense projection: Y[N,64] = A[N,512] @ W[512,64]  (fp32, V_WMMA_F32_16X16X4_F32)
// Block = 256 threads = 8 waves; each wave computes a 16x64 strip.
// Weight chunk is stored K-transposed in LDS as v2f fragments:
//   fragment(step s, half h, col n) = {W[kc+4s+2h][n], W[kc+4s+2h+1][n]}
//   at dword offset  HBASEh + n*TS + 2*s      -> one conflict-free ds_load_b64
// ---------------------------------------------------------------------------
__global__ __launch_bounds__(256) void ngcn_gemm_kernel(
    const float* __restrict__ A, const float* __restrict__ W,
    float* __restrict__ Y, int n_rows, int mtiles)
{
    __shared__ float wlds[HBASE1 + HALF_SZ];   // ~33.9 KB

    const int tid  = threadIdx.x;
    const int wave = tid >> 5;
    const int lane = tid & 31;
    const int lh   = lane >> 4;   // half-wave select (K split)
    const int l16  = lane & 15;

    int gt = blockIdx.x * 8 + wave;
    const bool live = (gt < mtiles);
    if (!live) gt = mtiles - 1;                 // clamp: keep EXEC all-ones for WMMA
    const long m0 = (long)gt * 16;

    // A-fragment source row for this lane (clamped against N for partial tiles)
    long arow = m0 + l16;
    if (arow >= n_rows) arow = n_rows - 1;
    const float* __restrict__ Ap = A + arow * IN_CH + 2 * lh;

    // per-lane LDS read base for B fragments
    const float* __restrict__ bb = wlds + (lh ? HBASE1 : HBASE0) + l16 * TS;

    v8f acc0 = {}, acc1 = {}, acc2 = {}, acc3 = {};

    for (int kc = 0; kc < IN_CH; kc += KC) {
        // ---- stage weight chunk rows [kc, kc+KC) K-transposed into LDS ----
        const float4* __restrict__ wsrc = (const float4*)(W + (long)kc * OUT_CH);
        #pragma unroll
        for (int i = 0; i < (KC * OUT_CH / 4) / 256; ++i) {
            const int j  = tid + i * 256;       // float4 index within chunk
            const int r  = j >> 4;              // chunk-local K row
            const int n0 = (j & 15) * 4;        // first output column
            const float4 v = wsrc[j];
            const int base = ((r & 2) ? HBASE1 : HBASE0) // half by (k mod 4) >= 2
                           + 2 * (r >> 2)                // pair slot s = r/4
                           + (r & 1);                    // component within pair
            wlds[base + (n0 + 0) * TS] = v.x;
            wlds[base + (n0 + 1) * TS] = v.y;
            wlds[base + (n0 + 2) * TS] = v.z;
            wlds[base + (n0 + 3) * TS] = v.w;
        }
        __syncthreads();

        // ---- 32 K-steps of 4, A fragment software-pipelined one step ahead ----
        v2f a_cur = *(const v2f*)(Ap + kc);
        #pragma unroll
        for (int s = 0; s < KC / 4; ++s) {
            const int sn = (s + 1) & (KC / 4 - 1);        // wrap: dummy reload on last
            const v2f a_nxt = *(const v2f*)(Ap + kc + sn * 4);

            const v2f b0 = *(const v2f*)(bb + 0 * 16 * TS + 2 * s);
            const v2f b1 = *(const v2f*)(bb + 1 * 16 * TS + 2 * s);
            const v2f b2 = *(const v2f*)(bb + 2 * 16 * TS + 2 * s);
            const v2f b3 = *(const v2f*)(bb + 3 * 16 * TS + 2 * s);

            acc0 = __builtin_amdgcn_wmma_f32_16x16x4_f32(false, a_cur, false, b0, (short)0, acc0, false, false);
            acc1 = __builtin_amdgcn_wmma_f32_16x16x4_f32(false, a_cur, false, b1, (short)0, acc1, false, false);
            acc2 = __builtin_amdgcn_wmma_f32_16x16x4_f32(false, a_cur, false, b2, (short)0, acc2, false, false);
            acc3 = __builtin_amdgcn_wmma_f32_16x16x4_f32(false, a_cur, false, b3, (short)0, acc3, false, false);
            a_cur = a_nxt;
        }
        __syncthreads();
    }

    if (live) {
        // C/D layout: VGPR r -> rows m0+r (lanes 0-15) and m0+8+r (lanes 16-31), col = l16
        float* __restrict__ yb = Y + (m0 + lh * 8) * OUT_CH + l16;
        #pragma unroll
        for (int r = 0; r < 8; ++r) {
            const long rr = m0 + lh * 8 + r;
            if (rr < n_rows) {
                yb[r * OUT_CH + 0 ] = acc0[r];
                yb[r * OUT_CH + 16] = acc1[r];
                yb[r * OUT_CH + 32] = acc2[r];
                yb[r * OUT_CH + 48] = acc3[r];
            }
        }
    }
}

// ---------------------------------------------------------------------------
// COO SpMM scatter: out[row[e], :] += val[e] * x[col[e], :]
// 16 threads per edge, float4 channel quad each: b128 gather + 4 atomic f32 adds.
// ---------------------------------------------------------------------------
__global__ __launch_bounds__(256) void ngcn_spmm_kernel(
    const int* __restrict__ rows, const int* __restrict__ cols,
    const float* __restrict__ vals, const float* __restrict__ xin,
    float* __restrict__ xout, int n_edges)
{
    const long t = (long)blockIdx.x * blockDim.x + threadIdx.x;
    const long e = t >> 4;
    if (e >= n_edges) return;
    const int q = ((int)t & 15) * 4;

    const int   r = rows[e];
    const int   c = cols[e];
    const float w = vals[e];

    const float4 xv = *(const float4*)(xin + (long)c * OUT_CH + q);
    float* __restrict__ o = xout + (long)r * OUT_CH + q;
    atomicAdd(o + 0, w * xv.x);
    atomicAdd(o + 1, w * xv.y);
    atomicAdd(o + 2, w * xv.z);
    atomicAdd(o + 3, w * xv.w);
}

// ---------------------------------------------------------------------------
// Init helpers: zero-fill and bias broadcast (bias folded into final SpMM base)
// ---------------------------------------------------------------------------
__global__ __launch_bounds__(256) void ngcn_zero_kernel(float* __restrict__ p, long n)
{
    const long i = (long)blockIdx.x * blockDim.x + threadIdx.x;
    if (i < n) p[i] = 0.0f;
}

__global__ __launch_bounds__(256) void ngcn_bias_kernel(
    float* __restrict__ p, const float* __restrict__ bias, long n)
{
    const long i = (long)blockIdx.x * blockDim.x + threadIdx.x;
    if (i < n) p[i] = bias[i & (OUT_CH - 1)];
}

extern "C" void kernel_launch(void* const* d_in, const int* in_sizes, int n_in,
                              void* d_out, int out_size, void* d_ws, size_t ws_size,
                              hipStream_t stream)
{
    const int*   adj    = (const int*)d_in[0];       // [2, E] int32
    const float* vals   = (const float*)d_in[1];     // [E]
    const float* feat   = (const float*)d_in[2];     // [N, 512]
    const float* weight = (const float*)d_in[3];     // [512, 64]
    const float* bias   = (const float*)d_in[4];     // [1, 64]

    const int E = in_sizes[1];
    const int N = in_sizes[2] / IN_CH;
    const int* rows = adj;
    const int* cols = adj + E;

    float* ws0 = (float*)d_ws;                       // projection output  [N,64]
    float* ws1 = ws0 + (long)N * OUT_CH;             // first SpMM output  [N,64]
    float* out = (float*)d_out;                      // final              [N,64]

    const long nelems   = (long)N * OUT_CH;
    const int  mtiles   = (N + 15) / 16;
    const int  gemm_gr  = (mtiles + 7) / 8;
    const int  fill_gr  = (int)((nelems + 255) / 256);
    const int  spmm_gr  = (int)(((long)E * 16 + 255) / 256);

    // 1) projection: ws0 = feat @ weight            (WMMA f32)
    ngcn_gemm_kernel<<<gemm_gr, 256, 0, stream>>>(feat, weight, ws0, N, mtiles);
    // 2) ws1 = A @ ws0
    ngcn_zero_kernel<<<fill_gr, 256, 0, stream>>>(ws1, nelems);
    ngcn_spmm_kernel<<<spmm_gr, 256, 0, stream>>>(rows, cols, vals, ws0, ws1, E);
    // 3) out = A @ ws1 + bias   (bias as the accumulation base)
    ngcn_bias_kernel<<<fill_gr, 256, 0, stream>>>(out, bias, nelems);
    ngcn_spmm_kernel<<<spmm_gr, 256, 0, stream>>>(rows, cols, vals, ws1, out, E);
}